// NBVLoss_88450556494510
// MI455X (gfx1250) — compile-verified
//
#include <hip/hip_runtime.h>

typedef __attribute__((ext_vector_type(4))) float f32x4;
typedef __attribute__((ext_vector_type(2))) float v2f;
typedef __attribute__((ext_vector_type(8))) float v8f;

#define NBLOCKS 2048
#define NTHREADS 256

// Per-element BCE with torch clamp semantics and bucket weights folded in.
// target is exactly 0.0f or 1.0f, so only one log term is live per element.
__device__ __forceinline__ float bce_elem(float p, float t) {
    const bool  one = (t != 0.0f);
    const float x   = one ? p : (1.0f - p);
    // ln(x) = log2(x) * ln(2); __log2f lowers to v_log_f32 (native trans op)
    float l = __log2f(x) * 0.69314718055994530942f;
    l = fmaxf(l, -100.0f);                 // torch BCELoss log clamp
    const float w = one ? 10.0f : 1.0f;    // LAMBDA_FOR1 / LAMBDA_FOR0
    return w * (-l);
}

__global__ void __launch_bounds__(NTHREADS)
bce_partial_kernel(const float* __restrict__ pred,
                   const float* __restrict__ targ,
                   int n4, int n,
                   float* __restrict__ partials) {
    const f32x4* p4 = (const f32x4*)pred;
    const f32x4* t4 = (const f32x4*)targ;

    float acc = 0.0f;
    const int idx    = blockIdx.x * NTHREADS + threadIdx.x;
    const int stride = gridDim.x * NTHREADS;

    // Streaming phase: 128-bit non-temporal loads (global_load_b128 TH=NT),
    // register-only accumulation. This is the bandwidth-critical path.
    for (int i = idx; i < n4; i += stride) {
        f32x4 p = __builtin_nontemporal_load(p4 + i);
        f32x4 t = __builtin_nontemporal_load(t4 + i);
        acc += bce_elem(p.x, t.x);
        acc += bce_elem(p.y, t.y);
        acc += bce_elem(p.z, t.z);
        acc += bce_elem(p.w, t.w);
    }
    // Scalar tail (n is a multiple of 4 for the reference shape; kept for safety).
    if (idx == 0) {
        for (int i = n4 * 4; i < n; ++i) acc += bce_elem(pred[i], targ[i]);
    }

    __shared__ float s[NTHREADS];
    s[threadIdx.x] = acc;
    __syncthreads();

    // Block reduction on the matrix pipe: wave 0 folds all 256 partials with
    // four chained V_WMMA_F32_16X16X4_F32 ops. A = all-ones 16x4, so
    // D[m][n] = sum_k B[k][n] + C[m][n]: every B element contributes exactly
    // once, and D is uniform across rows. Branch is wave-uniform -> EXEC is
    // all ones inside the WMMA region (required by the ISA).
    if (threadIdx.x < 32) {
        const int l = threadIdx.x;
        const v2f a = {1.0f, 1.0f};      // ones A-matrix (2 VGPRs/lane)
        v8f c = {};                      // f32 accumulator (8 VGPRs/lane)
#pragma unroll
        for (int j = 0; j < 4; ++j) {
            v2f b = { s[j * 64 + 2 * l], s[j * 64 + 2 * l + 1] };  // 64 partials as 4x16 B
            c = __builtin_amdgcn_wmma_f32_16x16x4_f32(
                    /*neg_a=*/false, a, /*neg_b=*/false, b,
                    /*c_mod=*/(short)0, c, /*reuse_a=*/false, /*reuse_b=*/false);
        }
        // c[0]: lane L holds column-sum[L % 16] (rows uniform). Sum the 16
        // distinct column sums with xor-shuffles within each 16-lane half.
        float v = c[0];
        v += __shfl_xor(v, 1, 32);
        v += __shfl_xor(v, 2, 32);
        v += __shfl_xor(v, 4, 32);
        v += __shfl_xor(v, 8, 32);
        if (l == 0) partials[blockIdx.x] = v;
    }
}

__global__ void __launch_bounds__(NTHREADS)
bce_final_kernel(const float* __restrict__ partials, int nb, float* __restrict__ out) {
    __shared__ float s[NTHREADS];
    float acc = 0.0f;
    for (int i = threadIdx.x; i < nb; i += NTHREADS) acc += partials[i];
    s[threadIdx.x] = acc;
    __syncthreads();
#pragma unroll
    for (int off = NTHREADS / 2; off > 0; off >>= 1) {
        if (threadIdx.x < off) s[threadIdx.x] += s[threadIdx.x + off];
        __syncthreads();
    }
    if (threadIdx.x == 0) out[0] = s[0];
}

extern "C" void kernel_launch(void* const* d_in, const int* in_sizes, int n_in,
                              void* d_out, int out_size, void* d_ws, size_t ws_size,
                              hipStream_t stream) {
    const float* pred = (const float*)d_in[0];
    const float* targ = (const float*)d_in[1];
    float* out = (float*)d_out;
    float* partials = (float*)d_ws;   // NBLOCKS floats of scratch (8 KB)

    const int n  = in_sizes[0];       // 4096*32768 = 2^27, fits in int
    const int n4 = n / 4;

    bce_partial_kernel<<<NBLOCKS, NTHREADS, 0, stream>>>(pred, targ, n4, n, partials);
    bce_final_kernel<<<1, NTHREADS, 0, stream>>>(partials, NBLOCKS, out);
}